// ResAdd_41480794144828
// MI455X (gfx1250) — compile-verified
//
#include <hip/hip_runtime.h>
#include <stdint.h>

#define B_    8
#define CIN   256
#define CMID  64
#define HW    784          // 28*28
#define NPIX  (B_*HW)      // 6272

// ---- workspace layout (float offsets) ----
#define WS_W1T   0                    // [c*64+o]           64*256
#define WS_W2T   16384                // [(c*9+tap)*64+o]   64*9*64
#define WS_W3T   53248                // [c*256+o]          64*256
#define WS_INV1  69632
#define WS_BETA1 69696
#define WS_INV2  69760
#define WS_BETA2 69824
#define WS_INV3  69888
#define WS_BETA3 70144
#define WS_H1    70400                // 8*64*784
#define WS_H2    (WS_H1 + 401408)
#define WS_H3    (WS_H1 + 2*401408)
// total ≈ 1.27M floats ≈ 4.9 MB

// ---- gfx1250 async global->LDS path (compile-guarded) ----
#if defined(__HIP_DEVICE_COMPILE__) && defined(__gfx1250__) && \
    __has_builtin(__builtin_amdgcn_global_load_async_to_lds_b32) && \
    __has_builtin(__builtin_amdgcn_s_wait_asynccnt)
#define USE_ASYNC 1
#else
#define USE_ASYNC 0
#endif

#if USE_ASYNC
// builtin signature (from hipcc diagnostic): arg0 is AS1 int*, arg1 is AS3 int*
typedef __attribute__((address_space(1))) int* gas_p;
typedef __attribute__((address_space(3))) int* las_p;
static __device__ __forceinline__ gas_p as_g(const void* p) {
  return (gas_p)(unsigned long long)p;
}
static __device__ __forceinline__ las_p as_l(void* p) {
  // generic LDS pointer: low 32 bits are the LDS byte offset (aperture scheme)
  return (las_p)(unsigned)(unsigned long long)p;
}
#endif

// ================= K0: weight transposes + BN folding =================
__global__ __launch_bounds__(256) void k0_prep(
    const float* __restrict__ w1, const float* __restrict__ w2, const float* __restrict__ w3,
    const float* __restrict__ g1, const float* __restrict__ b1, const float* __restrict__ m1, const float* __restrict__ v1,
    const float* __restrict__ g2, const float* __restrict__ b2, const float* __restrict__ m2, const float* __restrict__ v2,
    const float* __restrict__ g3, const float* __restrict__ b3, const float* __restrict__ m3, const float* __restrict__ v3,
    float* __restrict__ ws) {
  const int i = blockIdx.x * 256 + threadIdx.x;
  if (i < 16384) {               // w1t[c*64+o] = w1[o,c]
    int c = i >> 6, o = i & 63;
    ws[WS_W1T + i] = w1[o * CIN + c];
  }
  if (i < 36864) {               // w2t[(c*9+tap)*64+o] = w2[o,c,kh,kw]
    int o = i & 63, r = i >> 6;
    int tap = r % 9, c = r / 9;
    ws[WS_W2T + i] = w2[(o * 64 + c) * 9 + tap];
  }
  if (i < 16384) {               // w3t[c*256+o] = w3[o,c]
    int c = i >> 8, o = i & 255;
    ws[WS_W3T + i] = w3[o * 64 + c];
  }
  if (i < 64) {
    float inv = g1[i] / sqrtf(v1[i] + 1e-5f);
    ws[WS_INV1 + i] = inv; ws[WS_BETA1 + i] = b1[i] - m1[i] * inv;
  }
  if (i < 64) {
    float inv = g2[i] / sqrtf(v2[i] + 1e-5f);
    ws[WS_INV2 + i] = inv; ws[WS_BETA2 + i] = b2[i] - m2[i] * inv;
  }
  if (i < 256) {
    float inv = g3[i] / sqrtf(v3[i] + 1e-5f);
    ws[WS_INV3 + i] = inv; ws[WS_BETA3 + i] = b3[i] - m3[i] * inv;
  }
}

// ================= K1: 1x1 adder conv 256->64 =================
// grid (25 pixel-tiles, 2 cout-halves), block 256. Each lane: 1 pixel, 32 couts.
// Async double-buffered LDS staging of 32-channel x chunks (ASYNCcnt pipeline).
__global__ __launch_bounds__(256) void k1_adder1x1(
    const float* __restrict__ x, const float* __restrict__ w1t, float* __restrict__ h1) {
  const int t  = threadIdx.x;
  const int pg = blockIdx.x * 256 + t;
  const int pc = pg < NPIX ? pg : NPIX - 1;
  const int b  = pc / HW;
  const int p  = pc - b * HW;
  const int co0 = blockIdx.y * 32;
  const float* xb = x + (size_t)b * CIN * HW + p;   // + c*HW

  float acc[32];
#pragma unroll
  for (int o = 0; o < 32; ++o) acc[o] = 0.f;

#if USE_ASYNC
  __shared__ float tile[2][32 * 256];
#pragma unroll
  for (int i = 0; i < 32; ++i)
    __builtin_amdgcn_global_load_async_to_lds_b32(
        as_g(xb + i * HW), as_l(&tile[0][i * 256 + t]), 0, 0);
  for (int ch = 0; ch < 8; ++ch) {
    if (ch < 7) {
      const float* xs = xb + (ch + 1) * 32 * HW;
      float* ls = &tile[(ch + 1) & 1][0];
#pragma unroll
      for (int i = 0; i < 32; ++i)
        __builtin_amdgcn_global_load_async_to_lds_b32(
            as_g(xs + i * HW), as_l(ls + i * 256 + t), 0, 0);
      __builtin_amdgcn_s_wait_asynccnt(32);   // chunk ch complete (in-order)
    } else {
      __builtin_amdgcn_s_wait_asynccnt(0);
    }
    asm volatile("" ::: "memory");            // LDS now holds staged data
    const float* cw = w1t + ch * 32 * 64 + co0;   // uniform -> scalar loads
    const float* lt = &tile[ch & 1][0];
    for (int i = 0; i < 32; ++i) {
      float xv = lt[i * 256 + t];
#pragma unroll
      for (int o = 0; o < 32; ++o)
        acc[o] -= __builtin_fabsf(xv - cw[i * 64 + o]);
    }
  }
#else
  for (int c = 0; c < CIN; ++c) {
    float xv = xb[c * HW];
    const float* cw = w1t + c * 64 + co0;
#pragma unroll
    for (int o = 0; o < 32; ++o)
      acc[o] -= __builtin_fabsf(xv - cw[o]);
  }
#endif

  if (pg < NPIX) {
    float* hp = h1 + (size_t)b * CMID * HW + p;
#pragma unroll
    for (int o = 0; o < 32; ++o) hp[(co0 + o) * HW] = acc[o];
  }
}

// ================= K2: depthwise 3x3 adder (PEG) + BN1 + ReLU =================
// grid = 512 planes (b*64+c), block 256. Zero-padded taps still contribute -|0-w|.
__global__ __launch_bounds__(256) void k2_peg(
    const float* __restrict__ h1, const float* __restrict__ wp,
    const float* __restrict__ ws, float* __restrict__ h2) {
  const int plane = blockIdx.x;          // b*64 + c
  const int c = plane & 63;
  const float* in = h1 + (size_t)plane * HW;
  float wl[9];
#pragma unroll
  for (int k = 0; k < 9; ++k) wl[k] = wp[c * 9 + k];   // uniform
  const float s_inv  = ws[WS_INV1 + c];
  const float s_beta = ws[WS_BETA1 + c];

  for (int p = threadIdx.x; p < HW; p += 256) {
    const int y  = p / 28;
    const int x0 = p - y * 28;
    float acc = 0.f;
#pragma unroll
    for (int kh = 0; kh < 3; ++kh) {
      const int yy = y + kh - 1;
#pragma unroll
      for (int kw = 0; kw < 3; ++kw) {
        const int xc = x0 + kw - 1;
        const bool ib = ((unsigned)yy < 28u) & ((unsigned)xc < 28u);
        const float v = ib ? in[yy * 28 + xc] : 0.f;
        acc -= __builtin_fabsf(v - wl[kh * 3 + kw]);
      }
    }
    h2[(size_t)plane * HW + p] = fmaxf(acc * s_inv + s_beta, 0.f);
  }
}

// ================= K3: 3x3 adder conv 64->64 (pad=1) + BN2 + ReLU =================
// grid (25 pixel-tiles, 4 cout-tiles), block 256. Each lane: 1 pixel, 16 couts.
__global__ __launch_bounds__(256) void k3_adder3x3(
    const float* __restrict__ h2, const float* __restrict__ ws, float* __restrict__ h3) {
  const int t  = threadIdx.x;
  const int pg = blockIdx.x * 256 + t;
  const int pc = pg < NPIX ? pg : NPIX - 1;
  const int b  = pc / HW;
  const int p  = pc - b * HW;
  const int y  = p / 28;
  const int x0 = p - y * 28;
  const int co0 = blockIdx.y * 16;
  const float* in  = h2 + (size_t)b * CMID * HW;
  const float* w2t = ws + WS_W2T;

  float acc[16];
#pragma unroll
  for (int o = 0; o < 16; ++o) acc[o] = 0.f;

  for (int c = 0; c < CMID; ++c) {
    const float* ip = in + c * HW;
    if (c + 1 < CMID) __builtin_prefetch(ip + HW, 0, 0);  // next channel plane
    const float* cw = w2t + c * 9 * 64 + co0;
#pragma unroll
    for (int kh = 0; kh < 3; ++kh) {
      const int yy = y + kh - 1;
#pragma unroll
      for (int kw = 0; kw < 3; ++kw) {
        const int xc = x0 + kw - 1;
        const bool ib = ((unsigned)yy < 28u) & ((unsigned)xc < 28u);
        const float v = ib ? ip[yy * 28 + xc] : 0.f;
        const float* w = cw + (kh * 3 + kw) * 64;        // uniform 16 dwords
#pragma unroll
        for (int o = 0; o < 16; ++o)
          acc[o] -= __builtin_fabsf(v - w[o]);
      }
    }
  }

  if (pg < NPIX) {
    float* op = h3 + (size_t)b * CMID * HW + p;
#pragma unroll
    for (int o = 0; o < 16; ++o) {
      float r = acc[o] * ws[WS_INV2 + co0 + o] + ws[WS_BETA2 + co0 + o];
      op[(co0 + o) * HW] = fmaxf(r, 0.f);
    }
  }
}

// ================= K4: 1x1 adder conv 64->256 + BN3 + residual + ReLU =================
// grid (25 pixel-tiles, 16 cout-tiles), block 256. Each lane: 1 pixel, 16 couts.
__global__ __launch_bounds__(256) void k4_adder1x1_res(
    const float* __restrict__ h3, const float* __restrict__ ws,
    const float* __restrict__ x, float* __restrict__ out) {
  const int t  = threadIdx.x;
  const int pg = blockIdx.x * 256 + t;
  const int pc = pg < NPIX ? pg : NPIX - 1;
  const int b  = pc / HW;
  const int p  = pc - b * HW;
  const int co0 = blockIdx.y * 16;
  const float* ip  = h3 + (size_t)b * CMID * HW + p;
  const float* w3t = ws + WS_W3T;

  float acc[16];
#pragma unroll
  for (int o = 0; o < 16; ++o) acc[o] = 0.f;

  for (int c = 0; c < CMID; ++c) {
    const float v = ip[c * HW];
    const float* w = w3t + c * 256 + co0;                // uniform 16 dwords
#pragma unroll
    for (int o = 0; o < 16; ++o)
      acc[o] -= __builtin_fabsf(v - w[o]);
  }

  if (pg < NPIX) {
    const float* rx = x  + (size_t)b * CIN * HW + p;
    float*       op = out + (size_t)b * CIN * HW + p;
#pragma unroll
    for (int o = 0; o < 16; ++o) {
      float r = acc[o] * ws[WS_INV3 + co0 + o] + ws[WS_BETA3 + co0 + o]
              + rx[(co0 + o) * HW];
      op[(co0 + o) * HW] = fmaxf(r, 0.f);
    }
  }
}

// ================= host launch =================
extern "C" void kernel_launch(void* const* d_in, const int* in_sizes, int n_in,
                              void* d_out, int out_size, void* d_ws, size_t ws_size,
                              hipStream_t stream) {
  (void)in_sizes; (void)n_in; (void)out_size; (void)ws_size;
  const float* x    = (const float*)d_in[0];
  const float* w1   = (const float*)d_in[1];
  const float* wpeg = (const float*)d_in[2];
  const float* w2   = (const float*)d_in[3];
  const float* w3   = (const float*)d_in[4];
  const float* g1 = (const float*)d_in[5],  *b1 = (const float*)d_in[6];
  const float* m1 = (const float*)d_in[7],  *v1 = (const float*)d_in[8];
  const float* g2 = (const float*)d_in[9],  *b2 = (const float*)d_in[10];
  const float* m2 = (const float*)d_in[11], *v2 = (const float*)d_in[12];
  const float* g3 = (const float*)d_in[13], *b3 = (const float*)d_in[14];
  const float* m3 = (const float*)d_in[15], *v3 = (const float*)d_in[16];
  float* ws  = (float*)d_ws;
  float* out = (float*)d_out;

  float* h1 = ws + WS_H1;
  float* h2 = ws + WS_H2;
  float* h3 = ws + WS_H3;

  k0_prep<<<dim3(144), dim3(256), 0, stream>>>(w1, w2, w3,
      g1, b1, m1, v1, g2, b2, m2, v2, g3, b3, m3, v3, ws);
  k1_adder1x1<<<dim3(25, 2),  dim3(256), 0, stream>>>(x, ws + WS_W1T, h1);
  k2_peg     <<<dim3(512),    dim3(256), 0, stream>>>(h1, wpeg, ws, h2);
  k3_adder3x3<<<dim3(25, 4),  dim3(256), 0, stream>>>(h2, ws, h3);
  k4_adder1x1_res<<<dim3(25, 16), dim3(256), 0, stream>>>(h3, ws, x, out);
}